// N2V_GCN_EdgeModel_44023414784047
// MI455X (gfx1250) — compile-verified
//
#include <hip/hip_runtime.h>
#include <hip/hip_bf16.h>

typedef __attribute__((ext_vector_type(2))) float v2f;
typedef __attribute__((ext_vector_type(8))) float v8f;

// ---------------------------------------------------------------------------
// Degree / normalization kernels
// ---------------------------------------------------------------------------
__global__ void k_init_deg(float* __restrict__ deg, int n) {
    int i = blockIdx.x * blockDim.x + threadIdx.x;
    if (i < n) deg[i] = 1.0f;  // self-loop
}

__global__ void k_edge_deg(const int* __restrict__ dst, float* __restrict__ deg, int e) {
    int i = blockIdx.x * blockDim.x + threadIdx.x;
    if (i < e) atomicAdd(&deg[dst[i]], 1.0f);
}

__global__ void k_rsqrt(const float* __restrict__ deg, float* __restrict__ dinv, int n) {
    int i = blockIdx.x * blockDim.x + threadIdx.x;
    if (i < n) dinv[i] = rsqrtf(deg[i]);
}

// ---------------------------------------------------------------------------
// One WMMA k-step against 4 N-tiles; B fragments from a global row-major W.
// ---------------------------------------------------------------------------
__device__ __forceinline__ void wmma_step_g(v8f& a0, v8f& a1, v8f& a2, v8f& a3,
                                            v2f a, const float* __restrict__ W,
                                            int kg, int m) {
    const float* w0 = W + kg * 64 + m;
    const float* w1 = w0 + 64;
    v2f b0 = {w0[0],  w1[0]};
    v2f b1 = {w0[16], w1[16]};
    v2f b2 = {w0[32], w1[32]};
    v2f b3 = {w0[48], w1[48]};
    a0 = __builtin_amdgcn_wmma_f32_16x16x4_f32(false, a, false, b0, (short)0, a0, false, false);
    a1 = __builtin_amdgcn_wmma_f32_16x16x4_f32(false, a, false, b1, (short)0, a1, false, false);
    a2 = __builtin_amdgcn_wmma_f32_16x16x4_f32(false, a, false, b2, (short)0, a2, false, false);
    a3 = __builtin_amdgcn_wmma_f32_16x16x4_f32(false, a, false, b3, (short)0, a3, false, false);
}

// ---------------------------------------------------------------------------
// WMMA fp32 GEMM:  Out[M,64] = A[M,K] @ W[K,64]   (K = 128 or 64)
// One wave computes a 16x64 tile with 4 v8f accumulators.
// W is staged in LDS in k-paired layout: sW[(k/2)*128 + n*2 + (k&1)],
// so each B fragment {W[k][n], W[k+1][n]} is a single ds_load_b64.
// ---------------------------------------------------------------------------
template <int K>
__global__ __launch_bounds__(256) void k_gemm_n64(const float* __restrict__ A,
                                                  const float* __restrict__ W,
                                                  float* __restrict__ Out, int M) {
    __shared__ float sW[K * 64];
    for (int i = threadIdx.x; i < K * 64; i += 256) {
        int k = i >> 6, n = i & 63;
        sW[(k >> 1) * 128 + n * 2 + (k & 1)] = W[i];
    }
    __syncthreads();

    const int lane = threadIdx.x & 31;
    const int wave = threadIdx.x >> 5;
    const int tile = blockIdx.x * 8 + wave;
    if (tile * 16 >= M) return;  // wave-uniform: EXEC stays all-1 for live waves

    const int m  = lane & 15;
    const int kh = lane >> 4;
    const float* arow = A + (size_t)(tile * 16 + m) * K;

    v8f acc0 = {}, acc1 = {}, acc2 = {}, acc3 = {};
    for (int kb = 0; kb < K; kb += 4) {
        const int k = kb + 2 * kh;                 // even
        v2f a = *(const v2f*)(arow + k);           // global_load_b64
        const float* wp = sW + (k >> 1) * 128 + m * 2;
        v2f b0 = *(const v2f*)(wp + 0);            // ds_load_b64 each
        v2f b1 = *(const v2f*)(wp + 32);
        v2f b2 = *(const v2f*)(wp + 64);
        v2f b3 = *(const v2f*)(wp + 96);
        acc0 = __builtin_amdgcn_wmma_f32_16x16x4_f32(false, a, false, b0, (short)0, acc0, false, false);
        acc1 = __builtin_amdgcn_wmma_f32_16x16x4_f32(false, a, false, b1, (short)0, acc1, false, false);
        acc2 = __builtin_amdgcn_wmma_f32_16x16x4_f32(false, a, false, b2, (short)0, acc2, false, false);
        acc3 = __builtin_amdgcn_wmma_f32_16x16x4_f32(false, a, false, b3, (short)0, acc3, false, false);
    }
    // C/D layout: VGPR r -> row (r + 8*kh), col = 16*t + (lane&15)
    for (int r = 0; r < 8; ++r) {
        float* o = Out + (size_t)(tile * 16 + r + 8 * kh) * 64 + m;
        o[0]  = acc0[r];
        o[16] = acc1[r];
        o[32] = acc2[r];
        o[48] = acc3[r];
    }
}

// ---------------------------------------------------------------------------
// GCN aggregation: out = segsum(h[src]*coef, dst) + h*(1/deg) + b ; then ReLU
// ---------------------------------------------------------------------------
__global__ void k_agg_init4(const float4* __restrict__ h4, const float* __restrict__ dinv,
                            const float4* __restrict__ b4, float4* __restrict__ out4,
                            int total4) {
    int i = blockIdx.x * blockDim.x + threadIdx.x;
    if (i >= total4) return;
    int node = i >> 4;
    float di = dinv[node];
    float s = di * di;
    float4 hv = h4[i];
    float4 bv = b4[i & 15];
    float4 o;
    o.x = hv.x * s + bv.x;
    o.y = hv.y * s + bv.y;
    o.z = hv.z * s + bv.z;
    o.w = hv.w * s + bv.w;
    out4[i] = o;
}

__global__ void k_agg_edge(const float* __restrict__ h, const int* __restrict__ src,
                           const int* __restrict__ dst, const float* __restrict__ dinv,
                           float* __restrict__ out, int e) {
    int i = blockIdx.x * blockDim.x + threadIdx.x;
    if (i >= e * 64) return;
    int eidx = i >> 6, c = i & 63;
    int s = src[eidx], d = dst[eidx];
    float coef = dinv[s] * dinv[d];
    atomicAdd(&out[(size_t)d * 64 + c], h[(size_t)s * 64 + c] * coef);
}

__global__ void k_relu4(float4* __restrict__ x, int total4) {
    int i = blockIdx.x * blockDim.x + threadIdx.x;
    if (i >= total4) return;
    float4 v = x[i];
    v.x = fmaxf(v.x, 0.0f);
    v.y = fmaxf(v.y, 0.0f);
    v.z = fmaxf(v.z, 0.0f);
    v.w = fmaxf(v.w, 0.0f);
    x[i] = v;
}

// ---------------------------------------------------------------------------
// Edge head: feat = [u, v, |u-v|, u*v] (P x 256); z = relu(feat@M1 + mb1);
// out = z@M2 + mb2.  64 pairs per block (4 waves x 16), u/v staged in LDS.
// K-loop split into 4 branch-free region loops.
// ---------------------------------------------------------------------------
__global__ __launch_bounds__(128) void k_edge_head(
    const float* __restrict__ H, const int* __restrict__ pa, const int* __restrict__ pb,
    const float* __restrict__ M1, const float* __restrict__ mb1,
    const float* __restrict__ M2, const float* __restrict__ mb2,
    float* __restrict__ out, int P) {
    __shared__ float sU[64 * 64];
    __shared__ float sV[64 * 64];

    const int pair0 = blockIdx.x * 64;
    for (int i = threadIdx.x; i < 64 * 64; i += 128) {
        int p = i >> 6, c = i & 63;
        int q = pair0 + p;
        int ia = (q < P) ? pa[q] : 0;
        int ib = (q < P) ? pb[q] : 0;
        sU[i] = H[(size_t)ia * 64 + c];
        sV[i] = H[(size_t)ib * 64 + c];
    }
    __syncthreads();

    const int lane = threadIdx.x & 31;
    const int wave = threadIdx.x >> 5;
    const int m  = lane & 15;
    const int kh = lane >> 4;
    const float* u = sU + (wave * 16 + m) * 64;
    const float* v = sV + (wave * 16 + m) * 64;

    v8f acc0 = {}, acc1 = {}, acc2 = {}, acc3 = {};

    // region 0: feat = u
    for (int kb = 0; kb < 64; kb += 4) {
        const int kl = kb + 2 * kh;                // even
        v2f a = *(const v2f*)(u + kl);             // ds_load_b64
        wmma_step_g(acc0, acc1, acc2, acc3, a, M1, kl, m);
    }
    // region 1: feat = v
    for (int kb = 0; kb < 64; kb += 4) {
        const int kl = kb + 2 * kh;
        v2f a = *(const v2f*)(v + kl);
        wmma_step_g(acc0, acc1, acc2, acc3, a, M1, 64 + kl, m);
    }
    // region 2: feat = |u - v|
    for (int kb = 0; kb < 64; kb += 4) {
        const int kl = kb + 2 * kh;
        v2f uu = *(const v2f*)(u + kl);
        v2f vv = *(const v2f*)(v + kl);
        v2f a;
        a.x = fabsf(uu.x - vv.x);
        a.y = fabsf(uu.y - vv.y);
        wmma_step_g(acc0, acc1, acc2, acc3, a, M1, 128 + kl, m);
    }
    // region 3: feat = u * v
    for (int kb = 0; kb < 64; kb += 4) {
        const int kl = kb + 2 * kh;
        v2f uu = *(const v2f*)(u + kl);
        v2f vv = *(const v2f*)(v + kl);
        v2f a;
        a.x = uu.x * vv.x;
        a.y = uu.y * vv.y;
        wmma_step_g(acc0, acc1, acc2, acc3, a, M1, 192 + kl, m);
    }

    // z = relu(acc + mb1); out = z . M2 + mb2 (reduce 64 cols held across 16 lanes)
    const float add2 = mb2[0];
    for (int r = 0; r < 8; ++r) {
        float z0 = fmaxf(acc0[r] + mb1[m],      0.0f);
        float z1 = fmaxf(acc1[r] + mb1[m + 16], 0.0f);
        float z2 = fmaxf(acc2[r] + mb1[m + 32], 0.0f);
        float z3 = fmaxf(acc3[r] + mb1[m + 48], 0.0f);
        float s = z0 * M2[m] + z1 * M2[m + 16] + z2 * M2[m + 32] + z3 * M2[m + 48];
        s += __shfl_xor(s, 1, 32);
        s += __shfl_xor(s, 2, 32);
        s += __shfl_xor(s, 4, 32);
        s += __shfl_xor(s, 8, 32);
        int q = pair0 + wave * 16 + r + 8 * kh;
        if (m == 0 && q < P) out[q] = s + add2;
    }
}

// ---------------------------------------------------------------------------
// Launch
// ---------------------------------------------------------------------------
extern "C" void kernel_launch(void* const* d_in, const int* in_sizes, int n_in,
                              void* d_out, int out_size, void* d_ws, size_t ws_size,
                              hipStream_t stream) {
    const float* x   = (const float*)d_in[0];
    const int*   gei = (const int*)d_in[1];
    const int*   ep  = (const int*)d_in[2];
    const float* W1  = (const float*)d_in[3];
    const float* b1  = (const float*)d_in[4];
    const float* W2  = (const float*)d_in[5];
    const float* b2  = (const float*)d_in[6];
    const float* M1  = (const float*)d_in[7];
    const float* mb1 = (const float*)d_in[8];
    const float* M2  = (const float*)d_in[9];
    const float* mb2 = (const float*)d_in[10];
    float* out = (float*)d_out;

    const int N = in_sizes[0] / 128;
    const int E = in_sizes[1] / 2;
    const int P = in_sizes[2] / 2;
    const int* src = gei;
    const int* dst = gei + E;
    const int* pa  = ep;
    const int* pb  = ep + P;

    float* ws   = (float*)d_ws;
    float* deg  = ws;
    float* dinv = ws + N;
    float* bufA = ws + 2 * (size_t)N;              // N x 64 (gemm out)
    float* bufB = bufA + (size_t)N * 64;           // N x 64 (agg/relu out)

    const int T = 256;
    const int nTot4 = N * 16;                      // float4 count for N x 64
    const int tiles = (N + 15) / 16;
    const int gemmBlocks = (tiles + 7) / 8;
    const long long eTot = (long long)E * 64;
    const int aggBlocks = (int)((eTot + T - 1) / T);

    // degree + rsqrt
    k_init_deg<<<(N + T - 1) / T, T, 0, stream>>>(deg, N);
    k_edge_deg<<<(E + T - 1) / T, T, 0, stream>>>(dst, deg, E);
    k_rsqrt<<<(N + T - 1) / T, T, 0, stream>>>(deg, dinv, N);

    // layer 1: h = relu(gcn(x, W1, b1))
    k_gemm_n64<128><<<gemmBlocks, 256, 0, stream>>>(x, W1, bufA, N);
    k_agg_init4<<<(nTot4 + T - 1) / T, T, 0, stream>>>((const float4*)bufA, dinv,
                                                       (const float4*)b1, (float4*)bufB, nTot4);
    k_agg_edge<<<aggBlocks, T, 0, stream>>>(bufA, src, dst, dinv, bufB, E);
    k_relu4<<<(nTot4 + T - 1) / T, T, 0, stream>>>((float4*)bufB, nTot4);

    // layer 2: h = relu(gcn(h, W2, b2))
    k_gemm_n64<64><<<gemmBlocks, 256, 0, stream>>>(bufB, W2, bufA, N);
    k_agg_init4<<<(nTot4 + T - 1) / T, T, 0, stream>>>((const float4*)bufA, dinv,
                                                       (const float4*)b2, (float4*)bufB, nTot4);
    k_agg_edge<<<aggBlocks, T, 0, stream>>>(bufA, src, dst, dinv, bufB, E);
    k_relu4<<<(nTot4 + T - 1) / T, T, 0, stream>>>((float4*)bufB, nTot4);

    // edge head
    k_edge_head<<<(P + 63) / 64, 128, 0, stream>>>(bufB, pa, pb, M1, mb1, M2, mb2, out, P);
}